// GCN_79156247266018
// MI455X (gfx1250) — compile-verified
//
#include <hip/hip_runtime.h>
#include <cstdint>

// Problem constants (match reference)
#define NN    50000
#define EE    1000000
#define RR    8
#define BB    30
#define DIN   64
#define H1    64
#define H2    64

typedef __attribute__((ext_vector_type(2))) float v2f;
typedef __attribute__((ext_vector_type(8))) float v8f;

__device__ __forceinline__ void atomicAddF(float* p, float v) {
    __hip_atomic_fetch_add(p, v, __ATOMIC_RELAXED, __HIP_MEMORY_SCOPE_AGENT);
}

// ---------------------------------------------------------------------------
// Zero-fill: agg2 [N*64], deg_out [N], deg_in [N]
// ---------------------------------------------------------------------------
__global__ void zero_kernel(float* agg2, float* deg_out, float* deg_in, int n) {
    const int total = n * 64 + 2 * n;
    for (int i = blockIdx.x * blockDim.x + threadIdx.x; i < total;
         i += gridDim.x * blockDim.x) {
        if (i < n * 64)          agg2[i] = 0.0f;
        else if (i < n * 64 + n) deg_out[i - n * 64] = 0.0f;
        else                     deg_in[i - n * 64 - n] = 0.0f;
    }
}

// ---------------------------------------------------------------------------
// Wrel[r,i,o] = sum_b comp[r,b] * basis[b,i,o]   (8*64*64 outputs, 30 MACs)
// ---------------------------------------------------------------------------
__global__ void wrel_kernel(const float* __restrict__ comp,
                            const float* __restrict__ basis,
                            float* __restrict__ wrel) {
    int idx = blockIdx.x * blockDim.x + threadIdx.x;
    if (idx >= RR * DIN * H1) return;
    int r  = idx >> 12;        // /4096
    int io = idx & 4095;
    float acc = 0.0f;
#pragma unroll
    for (int b = 0; b < BB; ++b)
        acc += comp[r * BB + b] * basis[b * (DIN * H1) + io];
    wrel[idx] = acc;
}

// ---------------------------------------------------------------------------
// Degree accumulation (float, clipped at use sites)
// ---------------------------------------------------------------------------
__global__ void degree_kernel(const long long* __restrict__ src,
                              const long long* __restrict__ dst,
                              float* __restrict__ deg_out,
                              float* __restrict__ deg_in, int e) {
    int i = blockIdx.x * blockDim.x + threadIdx.x;
    if (i >= e) return;
    atomicAddF(deg_out + (int)src[i], 1.0f);
    atomicAddF(deg_in  + (int)dst[i], 1.0f);
}

// ---------------------------------------------------------------------------
// WMMA GEMM #1: one wave computes a 16(node) x 64 tile for one of 9 weight
// matrices: w<8 -> xr[node, w, :], w==8 -> h = X@loopW + bias1.
// Uses V_WMMA_F32_16X16X4_F32, K accumulated 4 at a time (16 steps).
//
// ISA 7.12.2 layouts (wave32):
//   A 16x4 f32: lane m=l&15, hi=l>>4; vgpr0=A[m][2*hi], vgpr1=A[m][2*hi+1]
//   B 4x16:     vgpr0=B[2*hi][n=m],   vgpr1=B[2*hi+1][n=m]
//   C/D 16x16:  vgpr v -> D[v+8*hi][n=m]
// ---------------------------------------------------------------------------
__global__ void gemm1_kernel(const float* __restrict__ X,
                             const float* __restrict__ wrel,
                             const float* __restrict__ loopW,
                             const float* __restrict__ bias1,
                             float* __restrict__ xr,
                             float* __restrict__ h, int nTiles) {
    const int wave = (blockIdx.x * blockDim.x + threadIdx.x) >> 5;
    if (wave >= nTiles * 9) return;              // wave-uniform guard
    const int lane = threadIdx.x & 31;
    const int tile = wave / 9;
    const int w    = wave - tile * 9;
    const int row0 = tile * 16;
    const int m    = lane & 15;
    const int hi   = lane >> 4;

    const float* __restrict__ W = (w < 8) ? (wrel + w * (DIN * H1)) : loopW;
    const float* __restrict__ arow = X + (size_t)(row0 + m) * DIN + 2 * hi;

    v8f acc0 = {}, acc1 = {}, acc2 = {}, acc3 = {};
#pragma unroll
    for (int k0 = 0; k0 < DIN; k0 += 4) {
        v2f a = *(const v2f*)(arow + k0);
        const float* wk = W + (size_t)(k0 + 2 * hi) * H1 + m;
        v2f b0, b1, b2, b3;
        b0[0] = wk[0];   b0[1] = wk[H1];
        b1[0] = wk[16];  b1[1] = wk[H1 + 16];
        b2[0] = wk[32];  b2[1] = wk[H1 + 32];
        b3[0] = wk[48];  b3[1] = wk[H1 + 48];
        acc0 = __builtin_amdgcn_wmma_f32_16x16x4_f32(false, a, false, b0, (short)0, acc0, false, false);
        acc1 = __builtin_amdgcn_wmma_f32_16x16x4_f32(false, a, false, b1, (short)0, acc1, false, false);
        acc2 = __builtin_amdgcn_wmma_f32_16x16x4_f32(false, a, false, b2, (short)0, acc2, false, false);
        acc3 = __builtin_amdgcn_wmma_f32_16x16x4_f32(false, a, false, b3, (short)0, acc3, false, false);
    }

    if (w < 8) {
        // xr layout: [N][8][64]
        float* base = xr + (size_t)row0 * (RR * H1) + (size_t)w * H1;
#pragma unroll
        for (int v = 0; v < 8; ++v) {
            float* drow = base + (size_t)(v + 8 * hi) * (RR * H1);
            drow[m]      = acc0[v];
            drow[16 + m] = acc1[v];
            drow[32 + m] = acc2[v];
            drow[48 + m] = acc3[v];
        }
    } else {
        const float bb0 = bias1[m], bb1 = bias1[16 + m],
                    bb2 = bias1[32 + m], bb3 = bias1[48 + m];
        float* base = h + (size_t)row0 * H1;
#pragma unroll
        for (int v = 0; v < 8; ++v) {
            float* drow = base + (size_t)(v + 8 * hi) * H1;
            drow[m]      = acc0[v] + bb0;
            drow[16 + m] = acc1[v] + bb1;
            drow[32 + m] = acc2[v] + bb2;
            drow[48 + m] = acc3[v] + bb3;
        }
    }
}

// ---------------------------------------------------------------------------
// Edge pass 1: wave per edge. Gather xr[src, etype, :] (one coalesced 256B
// row), scatter-add into h[dst, :] via L2 float atomics.
// ---------------------------------------------------------------------------
__global__ void edge1_kernel(const long long* __restrict__ src,
                             const long long* __restrict__ dst,
                             const long long* __restrict__ etype,
                             const float* __restrict__ xr,
                             float* __restrict__ h, int e) {
    const int wid = (blockIdx.x * blockDim.x + threadIdx.x) >> 5;
    if (wid >= e) return;
    const int lane = threadIdx.x & 31;
    const long long s = src[wid];
    const long long d = dst[wid];
    const long long t = etype[wid];
    v2f msg = *(const v2f*)(xr + ((size_t)s * RR + (size_t)t) * H1 + lane * 2);
    float* hrow = h + (size_t)d * H1 + lane * 2;
    atomicAddF(hrow + 0, msg[0]);
    atomicAddF(hrow + 1, msg[1]);
}

// ---------------------------------------------------------------------------
// Edge pass 2: wave per edge. val = h[src,:]*rsqrt(max(deg_out[src],1)),
// scatter-add into agg2[dst,:].
// ---------------------------------------------------------------------------
__global__ void edge2_kernel(const long long* __restrict__ src,
                             const long long* __restrict__ dst,
                             const float* __restrict__ h,
                             const float* __restrict__ deg_out,
                             float* __restrict__ agg2, int e) {
    const int wid = (blockIdx.x * blockDim.x + threadIdx.x) >> 5;
    if (wid >= e) return;
    const int lane = threadIdx.x & 31;
    const long long s = src[wid];
    const long long d = dst[wid];
    const float sc = rsqrtf(fmaxf(deg_out[s], 1.0f));
    v2f hv = *(const v2f*)(h + (size_t)s * H1 + lane * 2);
    float* arow = agg2 + (size_t)d * H1 + lane * 2;
    atomicAddF(arow + 0, hv[0] * sc);
    atomicAddF(arow + 1, hv[1] * sc);
}

// ---------------------------------------------------------------------------
// WMMA GEMM #2: out = (agg2 * rsqrt(max(deg_in,1))[:,None]) @ W2 + bias2
// A-row scaling folded into the A-fragment load.
// ---------------------------------------------------------------------------
__global__ void gemm2_kernel(const float* __restrict__ agg2,
                             const float* __restrict__ deg_in,
                             const float* __restrict__ W2,
                             const float* __restrict__ bias2,
                             float* __restrict__ out, int nTiles) {
    const int wave = (blockIdx.x * blockDim.x + threadIdx.x) >> 5;
    if (wave >= nTiles) return;
    const int lane = threadIdx.x & 31;
    const int row0 = wave * 16;
    const int m    = lane & 15;
    const int hi   = lane >> 4;

    const float sc = rsqrtf(fmaxf(deg_in[row0 + m], 1.0f));
    const float* __restrict__ arow = agg2 + (size_t)(row0 + m) * H1 + 2 * hi;

    v8f acc0 = {}, acc1 = {}, acc2 = {}, acc3 = {};
#pragma unroll
    for (int k0 = 0; k0 < H1; k0 += 4) {
        v2f a = *(const v2f*)(arow + k0);
        a[0] *= sc; a[1] *= sc;
        const float* wk = W2 + (size_t)(k0 + 2 * hi) * H2 + m;
        v2f b0, b1, b2, b3;
        b0[0] = wk[0];   b0[1] = wk[H2];
        b1[0] = wk[16];  b1[1] = wk[H2 + 16];
        b2[0] = wk[32];  b2[1] = wk[H2 + 32];
        b3[0] = wk[48];  b3[1] = wk[H2 + 48];
        acc0 = __builtin_amdgcn_wmma_f32_16x16x4_f32(false, a, false, b0, (short)0, acc0, false, false);
        acc1 = __builtin_amdgcn_wmma_f32_16x16x4_f32(false, a, false, b1, (short)0, acc1, false, false);
        acc2 = __builtin_amdgcn_wmma_f32_16x16x4_f32(false, a, false, b2, (short)0, acc2, false, false);
        acc3 = __builtin_amdgcn_wmma_f32_16x16x4_f32(false, a, false, b3, (short)0, acc3, false, false);
    }

    const float bb0 = bias2[m], bb1 = bias2[16 + m],
                bb2 = bias2[32 + m], bb3 = bias2[48 + m];
    float* base = out + (size_t)row0 * H2;
#pragma unroll
    for (int v = 0; v < 8; ++v) {
        float* drow = base + (size_t)(v + 8 * hi) * H2;
        drow[m]      = acc0[v] + bb0;
        drow[16 + m] = acc1[v] + bb1;
        drow[32 + m] = acc2[v] + bb2;
        drow[48 + m] = acc3[v] + bb3;
    }
}

// ---------------------------------------------------------------------------
extern "C" void kernel_launch(void* const* d_in, const int* in_sizes, int n_in,
                              void* d_out, int out_size, void* d_ws, size_t ws_size,
                              hipStream_t stream) {
    const float*     node_features = (const float*)d_in[0];
    const long long* edge_index    = (const long long*)d_in[1]; // [2,E] int64
    // d_in[2] = edge_norm : unused by the reference
    const long long* edge_type     = (const long long*)d_in[3];
    const float*     basis         = (const float*)d_in[4];
    const float*     comp          = (const float*)d_in[5];
    const float*     loop_weight   = (const float*)d_in[6];
    const float*     bias1         = (const float*)d_in[7];
    const float*     weight2       = (const float*)d_in[8];
    const float*     bias2         = (const float*)d_in[9];
    float*           out           = (float*)d_out;

    const int n = in_sizes[0] / DIN;   // 50000
    const int e = in_sizes[3];         // 1000000
    const long long* src = edge_index;
    const long long* dst = edge_index + e;

    // Workspace layout (bytes)
    char* ws = (char*)d_ws;
    float* wrel    = (float*)(ws);                                   // 128 KB
    float* xr      = (float*)(ws + (size_t)131072);                  // N*8*64*4 = 102.4 MB
    float* h       = (float*)(ws + (size_t)131072 + (size_t)n * RR * H1 * 4);
    float* agg2    = (float*)((char*)h    + (size_t)n * H1 * 4);
    float* deg_out = (float*)((char*)agg2 + (size_t)n * H1 * 4);
    float* deg_in  = (float*)((char*)deg_out + (size_t)n * 4);

    const int nTiles = n / 16;  // 3125 (N is a multiple of 16)

    // 1. zero agg2 + degrees
    zero_kernel<<<2048, 256, 0, stream>>>(agg2, deg_out, deg_in, n);
    // 2. relation weights from bases
    wrel_kernel<<<(RR * DIN * H1 + 255) / 256, 256, 0, stream>>>(comp, basis, wrel);
    // 3. degrees
    degree_kernel<<<(e + 255) / 256, 256, 0, stream>>>(src, dst, deg_out, deg_in, e);
    // 4. xr = X @ Wrel[r] (r=0..7), h = X @ loopW + bias1   [WMMA]
    {
        int waves = nTiles * 9;
        gemm1_kernel<<<(waves + 7) / 8, 256, 0, stream>>>(node_features, wrel,
                                                          loop_weight, bias1, xr, h, nTiles);
    }
    // 5. h += segment_sum(xr[src, etype], dst)
    edge1_kernel<<<(e * 32 + 255) / 256, 256, 0, stream>>>(src, dst, edge_type, xr, h, e);
    // 6. agg2 = segment_sum(h[src]*rsqrt(deg_out[src]), dst)
    edge2_kernel<<<(e * 32 + 255) / 256, 256, 0, stream>>>(src, dst, h, deg_out, agg2, e);
    // 7. out = (agg2 * rsqrt(deg_in)) @ W2 + bias2   [WMMA]
    gemm2_kernel<<<(nTiles + 7) / 8, 256, 0, stream>>>(agg2, deg_in, weight2, bias2, out, nTiles);
}